// SSDBoxCoder_38405597561160
// MI455X (gfx1250) — compile-verified
//
#include <hip/hip_runtime.h>
#include <hip/hip_bf16.h>
#include <stdint.h>

// ---------------------------------------------------------------------------
// SSD box decode + per-class NMS for MI455X (gfx1250, wave32).
//
// Pipeline:
//   1) decode_kernel   : decode 5460 boxes + areas                (ws)
//   2) iou_mask_kernel : 16x16 tiles; area_i+area_j via
//                        V_WMMA_F32_16X16X4_F32 (rank-2 matmul),
//                        intersection via VALU; threshold test is
//                        division-free (inter > T*denom); store
//                        (iou>0.45) as a bitmask (3.8 MB, L2-res) (ws)
//   3) sort_nms_kernel : 20 workgroups (1 class / WGP); bitonic
//                        sort of 8192 (score,idx) pairs in LDS,
//                        then exact sequential NMS over the
//                        bitmask, keep-bits in LDS                (ws)
//   4) writeout_kernel : dets (20,5460,5) then keep (20,5460)     (d_out)
// ---------------------------------------------------------------------------

#define NBOX 5460
#define NPOW 8192          // next pow2 >= NBOX (bitonic sort width)
#define NCLS 20            // foreground classes
#define NCH  21
#define WSTR 344           // u16 words per bitmask row (>= ceil(5460/16)=342)
#define NROWPAD 5472       // mask rows padded to multiple of 16
#define SCORE_THRESH_F 0.6f
#define NMS_THRESH_F   0.45f

typedef __attribute__((ext_vector_type(2))) float v2f;
typedef __attribute__((ext_vector_type(8))) float v8f;

// ---------------------------- 1) decode ------------------------------------
__global__ void decode_kernel(const float* __restrict__ loc,
                              const float* __restrict__ db,
                              float* __restrict__ boxes,
                              float* __restrict__ area) {
  int n = blockIdx.x * blockDim.x + threadIdx.x;
  if (n >= NBOX) return;
  float lx = loc[n * 4 + 0], ly = loc[n * 4 + 1];
  float lw = loc[n * 4 + 2], lh = loc[n * 4 + 3];
  float cx = db[n * 4 + 0], cy = db[n * 4 + 1];
  float dw = db[n * 4 + 2], dh = db[n * 4 + 3];
  float x = lx * 0.1f * dw + cx;
  float y = ly * 0.1f * dh + cy;
  float w = __expf(lw * 0.1f) * dw;
  float h = __expf(lh * 0.1f) * dh;
  boxes[n * 4 + 0] = x - w * 0.5f;
  boxes[n * 4 + 1] = y - h * 0.5f;
  boxes[n * 4 + 2] = x + w * 0.5f;
  boxes[n * 4 + 3] = y + h * 0.5f;
  area[n] = w * h;
}

// ------------------- 2) IoU threshold bitmask (WMMA) -----------------------
// One wave per 16x16 (i,j) tile. The denominator term area_i + area_j for the
// whole tile is computed on the matrix pipe as a rank-2 f32 matmul:
//   A(16x4) = [area_i, 1, 0, 0],  B(4x16) = [1 ; area_j ; 0 ; 0]
//   D[m][n] = area_i[m] + area_j[n]        (exact f32)
// while the VALU computes the min/max intersection terms. The IoU threshold
// is evaluated division-free:  inter/(sum-inter) > T  <=>  inter > T*(sum-inter)
// (valid because the union sum-inter is strictly positive).
__global__ void __launch_bounds__(32)
iou_mask_kernel(const float* __restrict__ boxes,
                const float* __restrict__ area,
                uint16_t* __restrict__ mask) {
  __shared__ float4 bi[16];
  __shared__ float4 bj[16];
  __shared__ unsigned int rowbits[16];

  const int lane = threadIdx.x;
  const int i0 = blockIdx.x * 16;
  const int j0 = blockIdx.y * 16;
  const int half = lane >> 4;   // 0: lanes 0-15, 1: lanes 16-31
  const int lm = lane & 15;

  const int gi = min(i0 + lm, NBOX - 1);
  const int gj = min(j0 + lm, NBOX - 1);
  if (half == 0) {
    bi[lm] = ((const float4*)boxes)[gi];
  } else {
    bj[lm] = ((const float4*)boxes)[gj];
  }
  if (lane < 16) rowbits[lane] = 0u;

  // A fragment: lanes 0-15 hold row M=lm, K={0,1}; lanes 16-31 hold K={2,3}=0.
  // B fragment: lanes 0-15 hold col N=lm, K={0,1}; lanes 16-31 hold K={2,3}=0.
  v2f a, b;
  a.x = half ? 0.0f : area[gi];   // K0 column of A = area_i
  a.y = half ? 0.0f : 1.0f;       // K1 column of A = ones
  b.x = half ? 0.0f : 1.0f;       // K0 row of B = ones
  b.y = half ? 0.0f : area[gj];   // K1 row of B = area_j
  v8f cacc = {};
  // D = A x B + 0 : D[m][n] = area_i[m] + area_j[n]
  v8f s = __builtin_amdgcn_wmma_f32_16x16x4_f32(
      false, a, false, b, (short)0, cacc, false, false);

  __syncthreads();

  // C/D layout: VGPR k, lane l -> M = k + 8*(l>=16), N = l&15
#pragma unroll
  for (int k = 0; k < 8; ++k) {
    const int M = k + 8 * half;
    const int Nn = lm;
    float4 A = bi[M];
    float4 B = bj[Nn];
    float ox = fminf(A.z, B.z) - fmaxf(A.x, B.x);
    float oy = fminf(A.w, B.w) - fmaxf(A.y, B.y);
    ox = fmaxf(ox, 0.0f);
    oy = fmaxf(oy, 0.0f);
    float inter = ox * oy;
    float denom = s[k] - inter;          // s[k] = area_i + area_j from WMMA
    // division-free threshold: inter/denom > T  <=>  inter > T*denom
    bool pred = (inter > NMS_THRESH_F * denom) &&
                (i0 + M < NBOX) && (j0 + Nn < NBOX);
    if (pred) atomicOr(&rowbits[M], 1u << Nn);
  }
  __syncthreads();

  if (lane < 16) {
    // rows padded to NROWPAD, so this store is always in range
    mask[(i0 + lane) * WSTR + (j0 >> 4)] = (uint16_t)rowbits[lane];
  }
}

// ------------------ 3) per-class bitonic sort + NMS ------------------------
// One workgroup per class. 50 KB static LDS (needs CDNA5's 320 KB WGP LDS).
__global__ void __launch_bounds__(1024)
sort_nms_kernel(const float* __restrict__ cls,
                const uint16_t* __restrict__ mask,
                unsigned char* __restrict__ keepout) {
  __shared__ unsigned int keys[NPOW];      // 32 KB: score bits
  __shared__ unsigned short ids[NPOW];     // 16 KB: original box index
  __shared__ unsigned int keepb[NPOW / 32];   // 1 KB
  __shared__ unsigned int validb[NPOW / 32];  // 1 KB

  const int tid = threadIdx.x;
  const int c = blockIdx.x;  // fg class 0..19 -> channel c+1

  for (int t = tid; t < NPOW; t += 1024) {
    if (t < NBOX) {
      float sc = cls[t * NCH + (c + 1)];
      keys[t] = __float_as_uint(sc);  // scores in [0,1): bit-monotonic
      ids[t] = (unsigned short)t;
    } else {
      keys[t] = 0u;
      ids[t] = 0xFFFFu;  // padding sorts after any real entry
    }
  }
  for (int t = tid; t < NPOW / 32; t += 1024) keepb[t] = 0xFFFFFFFFu;
  __syncthreads();

  // Bitonic sort: descending by key, ties -> ascending id
  // (matches stable jnp.argsort(-scores)).
  for (int k = 2; k <= NPOW; k <<= 1) {
    for (int j = k >> 1; j > 0; j >>= 1) {
      for (int t = tid; t < NPOW; t += 1024) {
        int p = t ^ j;
        if (p > t) {
          unsigned ke = keys[t], kp = keys[p];
          unsigned short ie = ids[t], ip = ids[p];
          bool desc = ((t & k) == 0);
          bool eLater = (ke < kp) || (ke == kp && ie > ip);
          bool pLater = (kp < ke) || (kp == ke && ip > ie);
          if (desc ? eLater : pLater) {
            keys[t] = kp; keys[p] = ke;
            ids[t] = ip;  ids[p] = ie;
          }
        }
      }
      __syncthreads();
    }
  }

  // valid bits from sorted scores (one thread per 32-bit word)
  for (int w = tid; w < NPOW / 32; w += 1024) {
    unsigned vv = 0u;
    for (int b = 0; b < 32; ++b) {
      int t = w * 32 + b;
      bool v = (t < NBOX) && (ids[t] != 0xFFFFu) &&
               (__uint_as_float(keys[t]) > SCORE_THRESH_F);
      vv |= (v ? (1u << b) : 0u);
    }
    validb[w] = vv;
  }
  __syncthreads();

  // Exact sequential NMS over sorted order; all real entries occupy
  // positions [0, NBOX) after the descending sort.
  for (int i = 0; i < NBOX; ++i) {
    __syncthreads();  // publish previous iteration's keep-bit clears
    bool cur = (((keepb[i >> 5] >> (i & 31)) & 1u) != 0u) &&
               (((validb[i >> 5] >> (i & 31)) & 1u) != 0u);
    if (!cur) continue;  // uniform decision (same LDS reads in all threads)
    const uint16_t* row = mask + (int)ids[i] * WSTR;
    for (int t = i + 1 + tid; t < NBOX; t += 1024) {
      if ((keepb[t >> 5] >> (t & 31)) & 1u) {
        int bj = ids[t];
        if ((row[bj >> 4] >> (bj & 15)) & 1u)
          atomicAnd(&keepb[t >> 5], ~(1u << (t & 31)));
      }
    }
  }
  __syncthreads();

  // scatter keep back to original box order (positions [0,NBOX) cover every id)
  for (int t = tid; t < NBOX; t += 1024) {
    int id = ids[t];
    bool k = (((keepb[t >> 5] >> (t & 31)) & 1u) != 0u) &&
             (((validb[t >> 5] >> (t & 31)) & 1u) != 0u);
    keepout[c * NBOX + id] = k ? 1 : 0;
  }
}

// ---------------------------- 4) write-out ---------------------------------
__global__ void writeout_kernel(const float* __restrict__ boxes,
                                const float* __restrict__ cls,
                                const unsigned char* __restrict__ keep,
                                float* __restrict__ out) {
  int g = blockIdx.x * blockDim.x + threadIdx.x;
  if (g >= NCLS * NBOX) return;
  int c = g / NBOX;
  int n = g - c * NBOX;
  float k = keep[g] ? 1.0f : 0.0f;
  float s = cls[n * NCH + c + 1];
  float* det = out + (size_t)g * 5;
  det[0] = boxes[n * 4 + 0] * k;
  det[1] = boxes[n * 4 + 1] * k;
  det[2] = boxes[n * 4 + 2] * k;
  det[3] = boxes[n * 4 + 3] * k;
  det[4] = s * k;
  out[(size_t)NCLS * NBOX * 5 + g] = k;  // keep mask as 0/1 float
}

// ---------------------------------------------------------------------------
extern "C" void kernel_launch(void* const* d_in, const int* in_sizes, int n_in,
                              void* d_out, int out_size, void* d_ws, size_t ws_size,
                              hipStream_t stream) {
  (void)in_sizes; (void)n_in; (void)out_size; (void)ws_size;
  const float* loc = (const float*)d_in[0];
  const float* cls = (const float*)d_in[1];
  const float* db  = (const float*)d_in[2];
  float* out = (float*)d_out;

  char* ws = (char*)d_ws;
  size_t off = 0;
  auto walloc = [&](size_t bytes) -> void* {
    off = (off + 255) & ~(size_t)255;
    void* p = ws + off;
    off += bytes;
    return p;
  };
  float* boxes = (float*)walloc((size_t)NBOX * 4 * sizeof(float));
  float* area  = (float*)walloc((size_t)NBOX * sizeof(float));
  uint16_t* mask = (uint16_t*)walloc((size_t)NROWPAD * WSTR * sizeof(uint16_t));
  unsigned char* keep = (unsigned char*)walloc((size_t)NCLS * NBOX);

  decode_kernel<<<(NBOX + 255) / 256, 256, 0, stream>>>(loc, db, boxes, area);

  dim3 tiles((NBOX + 15) / 16, (NBOX + 15) / 16);
  iou_mask_kernel<<<tiles, 32, 0, stream>>>(boxes, area, mask);

  sort_nms_kernel<<<NCLS, 1024, 0, stream>>>(cls, mask, keep);

  writeout_kernel<<<(NCLS * NBOX + 255) / 256, 256, 0, stream>>>(boxes, cls, keep, out);
}